// GNNModel_21285857919231
// MI455X (gfx1250) — compile-verified
//
#include <hip/hip_runtime.h>

#define D 128

typedef float v2f __attribute__((ext_vector_type(2)));
typedef float v8f __attribute__((ext_vector_type(8)));

#if defined(__HIP_DEVICE_COMPILE__) && __has_builtin(__builtin_amdgcn_wmma_f32_16x16x4_f32)
#define HAVE_WMMA_F32 1
#endif

__device__ __forceinline__ v8f wmma_acc(v2f a, v2f b, v8f c) {
#if defined(HAVE_WMMA_F32)
  // 8 args: (neg_a, A, neg_b, B, c_mod, C, reuse_a, reuse_b)
  return __builtin_amdgcn_wmma_f32_16x16x4_f32(false, a, false, b, (short)0, c,
                                               false, false);
#else
  (void)a; (void)b;
  return c;
#endif
}

__global__ void k_zero(float* __restrict__ p, int n) {
  int i = blockIdx.x * blockDim.x + threadIdx.x;
  if (i < n) p[i] = 0.f;
}

__global__ void k_count(const int* __restrict__ dst, float* __restrict__ deg, int e) {
  int i = blockIdx.x * blockDim.x + threadIdx.x;
  if (i < e) {
    __hip_atomic_fetch_add(&deg[dst[i]], 1.0f, __ATOMIC_RELAXED,
                           __HIP_MEMORY_SCOPE_AGENT);
  }
}

__global__ void k_dinv(const float* __restrict__ deg, float* __restrict__ dinv, int n) {
  int i = blockIdx.x * blockDim.x + threadIdx.x;
  if (i < n) dinv[i] = rsqrtf(deg[i] + 1.0f);
}

// Fused: H = act(X) @ W  (act = optional ReLU on load), then epilogue writes
//   H[m][c]   (needed for edge gather)
//   AGG[m][c] = H[m][c] * dinv[m]^2 + bias[c]   (self-loop + bias init)
// One wave computes a 16x128 strip via V_WMMA_F32_16X16X4_F32, K = 128.
__global__ void __launch_bounds__(256) k_gcn_gemm(
    const float* __restrict__ X, const float* __restrict__ W,
    const float* __restrict__ bias, const float* __restrict__ dinv,
    float* __restrict__ H, float* __restrict__ AGG, int nrows, int relu_in) {
  const int wave = threadIdx.x >> 5;
  const int lane = threadIdx.x & 31;
  const int half = lane >> 4;   // which 16-lane half
  const int l16  = lane & 15;
  const int row0 = (blockIdx.x * 8 + wave) * 16;
  if (row0 >= nrows) return;  // wave-uniform; active waves keep EXEC all-ones

  const int m = row0 + l16;                 // A-matrix row owned by this lane
  const bool mvalid = (m < nrows);
  const float* __restrict__ arow = X + (size_t)(mvalid ? m : 0) * D;

  const v8f vzero = {0.f, 0.f, 0.f, 0.f, 0.f, 0.f, 0.f, 0.f};
  v8f acc[8];
#pragma unroll
  for (int i = 0; i < 8; ++i) acc[i] = vzero;

  for (int k0 = 0; k0 < D; k0 += 4) {
    const int ka = k0 + 2 * half;  // this lane-half supplies K = ka, ka+1
    float a0 = arow[ka];
    float a1 = arow[ka + 1];
    if (relu_in) { a0 = fmaxf(a0, 0.f); a1 = fmaxf(a1, 0.f); }
    if (!mvalid) { a0 = 0.f; a1 = 0.f; }
    v2f a; a.x = a0; a.y = a1;
    const float* __restrict__ w0 = W + (size_t)ka * D + l16;
    const float* __restrict__ w1 = w0 + D;
#pragma unroll
    for (int ct = 0; ct < 8; ++ct) {        // 8 column tiles of 16
      v2f b; b.x = w0[ct * 16]; b.y = w1[ct * 16];
      acc[ct] = wmma_acc(a, b, acc[ct]);
    }
  }

  // D layout: acc[ct][v] = C[M = v + 8*half][N = ct*16 + l16]
#pragma unroll
  for (int v = 0; v < 8; ++v) {
    const int mm = row0 + v + 8 * half;
    if (mm < nrows) {
      const float di = dinv[mm];
      const float s = di * di;
      const size_t base = (size_t)mm * D;
#pragma unroll
      for (int ct = 0; ct < 8; ++ct) {
        const int col = ct * 16 + l16;
        const float hv = acc[ct][v];
        H[base + col] = hv;
        AGG[base + col] = fmaf(hv, s, bias[col]);
      }
    }
  }
}

// One wave per edge: AGG[dst] += dinv[src]*dinv[dst] * H[src]  (128 floats,
// float4 per lane, 4 agent-scope f32 atomics -> global_atomic_add_f32, L2-resident)
__global__ void __launch_bounds__(256) k_scatter(
    const float* __restrict__ H, const float* __restrict__ dinv,
    const int* __restrict__ src, const int* __restrict__ dst,
    float* __restrict__ AGG, int nedges) {
  const int lane = threadIdx.x & 31;
  const int e = (int)(((size_t)blockIdx.x * blockDim.x + threadIdx.x) >> 5);
  if (e >= nedges) return;
  const int s = src[e];
  const int d = dst[e];
  const float coef = dinv[s] * dinv[d];
  const float4 hv = ((const float4*)(H + (size_t)s * D))[lane];
  float* __restrict__ ap = AGG + (size_t)d * D + lane * 4;
  __hip_atomic_fetch_add(ap + 0, hv.x * coef, __ATOMIC_RELAXED, __HIP_MEMORY_SCOPE_AGENT);
  __hip_atomic_fetch_add(ap + 1, hv.y * coef, __ATOMIC_RELAXED, __HIP_MEMORY_SCOPE_AGENT);
  __hip_atomic_fetch_add(ap + 2, hv.z * coef, __ATOMIC_RELAXED, __HIP_MEMORY_SCOPE_AGENT);
  __hip_atomic_fetch_add(ap + 3, hv.w * coef, __ATOMIC_RELAXED, __HIP_MEMORY_SCOPE_AGENT);
}

extern "C" void kernel_launch(void* const* d_in, const int* in_sizes, int n_in,
                              void* d_out, int out_size, void* d_ws, size_t ws_size,
                              hipStream_t stream) {
  (void)n_in; (void)out_size; (void)ws_size;
  const float* x  = (const float*)d_in[0];
  const int*   ei = (const int*)d_in[1];
  const float* W1 = (const float*)d_in[2];
  const float* b1 = (const float*)d_in[3];
  const float* W2 = (const float*)d_in[4];
  const float* b2 = (const float*)d_in[5];
  const int n = in_sizes[0] / D;   // 50000
  const int e = in_sizes[1] / 2;   // 600000
  const int* srcv = ei;
  const int* dstv = ei + e;

  // Workspace layout (16B-aligned): dinv[n] | H[n*D] | AGG[n*D]
  char* ws = (char*)d_ws;
  float* dinv = (float*)ws;
  size_t off = (((size_t)n * 4) + 255) & ~(size_t)255;
  float* Hbuf = (float*)(ws + off);
  off += (((size_t)n * D * 4) + 255) & ~(size_t)255;
  float* Abuf = (float*)(ws + off);
  float* deg = Hbuf;  // alias: degree buffer only lives before first GEMM
  float* outp = (float*)d_out;

  const dim3 blk(256);
  const int nblk = (n + 255) / 256;
  const int eblk = (e + 255) / 256;
  const int gblk = (n + 127) / 128;   // 8 waves x 16 rows per block
  const int sblk = (e + 7) / 8;       // 8 edges (waves) per block

  // degrees -> dinv = rsqrt(deg + 1)
  k_zero <<<dim3(nblk), blk, 0, stream>>>(deg, n);
  k_count<<<dim3(eblk), blk, 0, stream>>>(dstv, deg, e);
  k_dinv <<<dim3(nblk), blk, 0, stream>>>(deg, dinv, n);

  // Layer 1: H = x@W1 ; AGG = H*dinv^2 + b1 ; AGG += scatter(H)
  k_gcn_gemm<<<dim3(gblk), blk, 0, stream>>>(x, W1, b1, dinv, Hbuf, Abuf, n, 0);
  k_scatter <<<dim3(sblk), blk, 0, stream>>>(Hbuf, dinv, srcv, dstv, Abuf, e);

  // Layer 2: H = relu(AGG)@W2 ; out = H*dinv^2 + b2 ; out += scatter(H)
  k_gcn_gemm<<<dim3(gblk), blk, 0, stream>>>(Abuf, W2, b2, dinv, Hbuf, outp, n, 1);
  k_scatter <<<dim3(sblk), blk, 0, stream>>>(Hbuf, dinv, srcv, dstv, outp, e);
}